// DynaMix_79044578116186
// MI455X (gfx1250) — compile-verified
//
#include <hip/hip_runtime.h>
#include <hip/hip_bf16.h>
#include <math.h>

// Problem constants (reference): M=128, N=64, E=32, P=2, S=512, B=8192
#define MM   128
#define NN   64
#define EE   32
#define PP   2
#define SS   512
#define BB   8192
#define MN   192          // M + N
#define BT   64           // K4 b-tile per workgroup
#define SBP  80           // LDS row stride (floats), padded: bank shift 16/row

typedef float v2f __attribute__((ext_vector_type(2)));
typedef float v4f __attribute__((ext_vector_type(4)));
typedef float v8f __attribute__((ext_vector_type(8)));

// ---------------------------------------------------------------------------
// K1a: z_cur[n,b] = (D @ z)[n,b] + sigma_g[n] * noise[n,b]        (N x B)
// ---------------------------------------------------------------------------
__global__ void k1_zcur(const float* __restrict__ D, const float* __restrict__ z,
                        const float* __restrict__ sigma_g, const float* __restrict__ noise,
                        float* __restrict__ z_cur) {
    int idx = blockIdx.x * blockDim.x + threadIdx.x;   // NN*BB threads
    int n = idx >> 13;           // / BB
    int b = idx & (BB - 1);
    float acc = sigma_g[n] * noise[idx];
    #pragma unroll 8
    for (int m = 0; m < MM; ++m)
        acc += D[n * MM + m] * z[m * BB + b];
    z_cur[idx] = acc;
}

// ---------------------------------------------------------------------------
// K1b: zcat[k,b] = k < M-P ? z[k,b] : relu(z[k,b])                (M x B)
// ---------------------------------------------------------------------------
__global__ void k1_zcat(const float* __restrict__ z, float* __restrict__ zcat) {
    int idx = blockIdx.x * blockDim.x + threadIdx.x;   // MM*BB threads
    int k = idx >> 13;
    float v = z[idx];
    if (k >= MM - PP) v = fmaxf(v, 0.0f);
    zcat[idx] = v;
}

// ---------------------------------------------------------------------------
// K2: single-pass context stream with ONLINE softmax.
// One wave32 per TWO batch columns: half-wave per b, lane holds 4 channels
// (16B global_load_b128, non-temporal: context is 1.07GB streamed once and
// must not evict L2-resident weights). Half-wave shuffle reduction (4 steps)
// for the L1 distance. Produces
//   c0[b,:] = sum_s attn[s,b]*context[s,b,:]
//   c1[b,:] = sum_s attn[s,b]*context[s+1,b,:]
// ---------------------------------------------------------------------------
__global__ void k2_stream(const float* __restrict__ ctx, const float* __restrict__ z_cur,
                          const float* __restrict__ temp1,
                          float* __restrict__ c0w, float* __restrict__ c1w) {
    int gid  = blockIdx.x * blockDim.x + threadIdx.x;
    int wid  = gid >> 5;
    int lane = gid & 31;
    int half = lane >> 4;          // which of the two b columns
    int q    = lane & 15;
    int b    = 2 * wid + half;
    int i0   = 4 * q;              // 4 channels per lane

    float zc0 = z_cur[(i0 + 0) * BB + b];
    float zc1 = z_cur[(i0 + 1) * BB + b];
    float zc2 = z_cur[(i0 + 2) * BB + b];
    float zc3 = z_cur[(i0 + 3) * BB + b];

    // softmax(-dist/|T|) in base-2: fold log2(e) into invT
    const float invT = 1.4426950408889634f / fabsf(temp1[0]);

    const v4f* p = (const v4f*)(ctx + (size_t)b * NN + i0);
    const size_t stride4 = (size_t)BB * NN / 4;   // s-stride in float4s

    v4f pr = *p;                                   // context[s] tile
    float mrun = -INFINITY, lrun = 0.0f;
    float c00 = 0.f, c01 = 0.f, c02 = 0.f, c03 = 0.f;
    float c10 = 0.f, c11 = 0.f, c12 = 0.f, c13 = 0.f;

    for (int s = 0; s < SS - 1; ++s) {
        const v4f* pn = p + stride4;
        v4f nx = __builtin_nontemporal_load(pn);         // context[s+1]
        __builtin_prefetch(pn + stride4, 0, 1);          // global_prefetch_b8

        // L1 distance over N=64 channels: 4/lane + 16-lane reduction
        float d = fabsf(pr.x - zc0) + fabsf(pr.y - zc1)
                + fabsf(pr.z - zc2) + fabsf(pr.w - zc3);
        #pragma unroll
        for (int off = 8; off > 0; off >>= 1)
            d += __shfl_xor(d, off, 32);                 // stays in half-wave

        float v     = -d * invT;
        float mnew  = fmaxf(mrun, v);
        float alpha = exp2f(mrun - mnew);
        float pw    = exp2f(v - mnew);
        lrun = lrun * alpha + pw;
        c00 = c00 * alpha + pw * pr.x;
        c01 = c01 * alpha + pw * pr.y;
        c02 = c02 * alpha + pw * pr.z;
        c03 = c03 * alpha + pw * pr.w;
        c10 = c10 * alpha + pw * nx.x;                   // c1 uses context[s+1]
        c11 = c11 * alpha + pw * nx.y;
        c12 = c12 * alpha + pw * nx.z;
        c13 = c13 * alpha + pw * nx.w;
        mrun = mnew;
        pr = nx;
        p = pn;
    }

    float inv_l = 1.0f / lrun;
    c0w[b * NN + i0 + 0] = c00 * inv_l;
    c0w[b * NN + i0 + 1] = c01 * inv_l;
    c0w[b * NN + i0 + 2] = c02 * inv_l;
    c0w[b * NN + i0 + 3] = c03 * inv_l;
    c1w[b * NN + i0 + 0] = c10 * inv_l;
    c1w[b * NN + i0 + 1] = c11 * inv_l;
    c1w[b * NN + i0 + 2] = c12 * inv_l;
    c1w[b * NN + i0 + 3] = c13 * inv_l;
}

// ---------------------------------------------------------------------------
// K3: emb -> MLP -> expert softmax.   One wave32 per b, 8 waves/block.
// ---------------------------------------------------------------------------
__global__ void k3_gate(const float* __restrict__ c0w, const float* __restrict__ c1w,
                        const float* __restrict__ zg,
                        const float* __restrict__ conv_w, const float* __restrict__ conv_b,
                        const float* __restrict__ W1, const float* __restrict__ b1,
                        const float* __restrict__ W2, const float* __restrict__ b2,
                        const float* __restrict__ temp2, float* __restrict__ w_expT) {
    __shared__ float sc0[8][NN];
    __shared__ float sc1[8][NN];
    __shared__ float semb[8][NN];
    __shared__ float sz[8][MM];

    int tid  = threadIdx.x;
    int wv   = tid >> 5;
    int lane = tid & 31;
    int b    = blockIdx.x * 8 + wv;

    sc0[wv][lane]      = c0w[b * NN + lane];
    sc0[wv][lane + 32] = c0w[b * NN + 32 + lane];
    sc1[wv][lane]      = c1w[b * NN + lane];
    sc1[wv][lane + 32] = c1w[b * NN + 32 + lane];
    #pragma unroll
    for (int r = 0; r < 4; ++r)
        sz[wv][lane + 32 * r] = zg[(lane + 32 * r) * BB + b];
    __syncthreads();

    #pragma unroll
    for (int t = 0; t < 2; ++t) {
        int n = lane + 32 * t;
        float acc = conv_b[n];
        #pragma unroll 8
        for (int i = 0; i < NN; ++i)
            acc += sc0[wv][i] * conv_w[n * 2 * NN + 2 * i]
                 + sc1[wv][i] * conv_w[n * 2 * NN + 2 * i + 1];
        semb[wv][n] = acc;
    }
    __syncthreads();

    float a1 = b1[lane];
    #pragma unroll 8
    for (int j = 0; j < NN; ++j) a1 += semb[wv][j] * W1[lane * MN + j];
    #pragma unroll 8
    for (int j = 0; j < MM; ++j) a1 += sz[wv][j] * W1[lane * MN + NN + j];
    a1 = fmaxf(a1, 0.0f);

    float a2 = b2[lane];
    #pragma unroll
    for (int f = 0; f < EE; ++f)
        a2 += __shfl(a1, f, 32) * W2[lane * EE + f];

    float v = -a2 / fabsf(temp2[0]);
    float mx = v;
    #pragma unroll
    for (int off = 16; off > 0; off >>= 1) mx = fmaxf(mx, __shfl_xor(mx, off, 32));
    float pr = __expf(v - mx);
    float sm = pr;
    #pragma unroll
    for (int off = 16; off > 0; off >>= 1) sm += __shfl_xor(sm, off, 32);

    w_expT[b * EE + lane] = pr / sm;
}

// ---------------------------------------------------------------------------
// K4: expert mixture via f32 WMMA (v_wmma_f32_16x16x4_f32).
// out[m,b] = z[m,b]*(sum_e w A[e,m]) + (sum_e w h[e,m])
//          + sum_e w_exp[b,e] * (Wx[e] @ zcat)[m,b]
// Block = 64-wide b-tile: zcat panel (128x64, padded stride 80) staged to LDS
// with gfx1250 ASYNC copies (global_load_async_to_lds_b128 + s_wait_asynccnt).
// Wave = one 16-row m-tile x 4 b-subtiles: each A-fragment feeds 4 WMMAs and
// Wx L2 traffic drops 4x vs a 16-wide tile.
// ---------------------------------------------------------------------------
__global__ void k4_experts(const float* __restrict__ Wx, const float* __restrict__ zcat,
                           const float* __restrict__ w_expT,
                           const float* __restrict__ A, const float* __restrict__ h,
                           const float* __restrict__ zg, float* __restrict__ out) {
    __shared__ float sB[MM][SBP];            // 128 x 80 floats = 40 KB

    int tid   = threadIdx.x;
    int wv    = tid >> 5;
    int lane  = tid & 31;
    int btile = blockIdx.x * BT;

    // --- async stage: zcat[:, btile:btile+64] -> LDS (16B chunks) ---------
    {
        unsigned int lds_base = (unsigned int)(unsigned long long)&sB[0][0];
        #pragma unroll
        for (int it = 0; it < (MM * (BT / 4)) / 256; ++it) {  // 2048 chunks / 256 thr
            int c = tid + 256 * it;
            int k = c >> 4;                  // row 0..127
            int j = c & 15;                  // 16B chunk within row
            unsigned int ldsa = lds_base + (unsigned int)((k * SBP + j * 4) * 4);
            unsigned long long ga =
                (unsigned long long)(const void*)(zcat + (size_t)k * BB + btile + 4 * j);
            asm volatile("global_load_async_to_lds_b128 %0, %1, off"
                         :: "v"(ldsa), "v"(ga) : "memory");
        }
        asm volatile("s_wait_asynccnt 0x0" ::: "memory");
    }
    __syncthreads();

    int mtile = wv * 16;
    int half  = lane >> 4;                   // 0: k-pair {0,1}; 1: {2,3}
    int l16   = lane & 15;

    v8f acc[4];
    #pragma unroll
    for (int j = 0; j < 4; ++j) acc[j] = (v8f){0.f,0.f,0.f,0.f,0.f,0.f,0.f,0.f};

    for (int e = 0; e < EE; ++e) {
        const float* wxe = Wx + (size_t)e * MM * MM + (mtile + l16) * MM;
        v8f g[4];
        #pragma unroll
        for (int j = 0; j < 4; ++j) g[j] = (v8f){0.f,0.f,0.f,0.f,0.f,0.f,0.f,0.f};

        #pragma unroll 4
        for (int kk = 0; kk < MM / 4; ++kk) {
            int kb = 4 * kk + 2 * half;
            v2f a;
            a.x = wxe[kb];
            a.y = wxe[kb + 1];
            #pragma unroll
            for (int j = 0; j < 4; ++j) {
                v2f bb;
                bb.x = sB[kb][16 * j + l16];
                bb.y = sB[kb + 1][16 * j + l16];
                g[j] = __builtin_amdgcn_wmma_f32_16x16x4_f32(
                           false, a, false, bb, (short)0, g[j], false, false);
            }
        }
        #pragma unroll
        for (int j = 0; j < 4; ++j) {
            float we = w_expT[(btile + 16 * j + l16) * EE + e];
            #pragma unroll
            for (int r = 0; r < 8; ++r) acc[j][r] += we * g[j][r];
        }
    }

    // Add z*Aw + hw terms and store. C/D layout: VGPR r -> m = mtile+r+8*half.
    #pragma unroll
    for (int j = 0; j < 4; ++j) {
        int col = btile + 16 * j + l16;
        float aw[8], hw[8];
        #pragma unroll
        for (int r = 0; r < 8; ++r) { aw[r] = 0.f; hw[r] = 0.f; }
        for (int e = 0; e < EE; ++e) {
            float we = w_expT[col * EE + e];
            #pragma unroll
            for (int r = 0; r < 8; ++r) {
                int m = mtile + r + 8 * half;
                aw[r] += we * A[e * MM + m];
                hw[r] += we * h[e * MM + m];
            }
        }
        #pragma unroll
        for (int r = 0; r < 8; ++r) {
            int m = mtile + r + 8 * half;
            out[m * BB + col] = zg[m * BB + col] * aw[r] + hw[r] + acc[j][r];
        }
    }
}

// ---------------------------------------------------------------------------
// Launch
// ---------------------------------------------------------------------------
extern "C" void kernel_launch(void* const* d_in, const int* in_sizes, int n_in,
                              void* d_out, int out_size, void* d_ws, size_t ws_size,
                              hipStream_t stream) {
    const float* z      = (const float*)d_in[0];
    const float* ctx    = (const float*)d_in[1];
    const float* noise  = (const float*)d_in[2];
    const float* conv_w = (const float*)d_in[3];
    const float* conv_b = (const float*)d_in[4];
    const float* D      = (const float*)d_in[5];
    const float* sig    = (const float*)d_in[6];
    const float* temp1  = (const float*)d_in[7];
    const float* W1     = (const float*)d_in[8];
    const float* b1     = (const float*)d_in[9];
    const float* W2     = (const float*)d_in[10];
    const float* b2     = (const float*)d_in[11];
    const float* temp2  = (const float*)d_in[12];
    const float* A      = (const float*)d_in[13];
    const float* Wx     = (const float*)d_in[14];
    const float* h      = (const float*)d_in[15];
    float* out = (float*)d_out;

    // workspace carve-up (floats), ~11.5 MB total
    float* ws     = (float*)d_ws;
    float* z_cur  = ws;                          // N*B
    float* c0w    = z_cur + NN * BB;             // B*N
    float* c1w    = c0w + BB * NN;               // B*N
    float* w_expT = c1w + BB * NN;               // B*E
    float* zcatw  = w_expT + BB * EE;            // M*B

    k1_zcur<<<(NN * BB) / 256, 256, 0, stream>>>(D, z, sig, noise, z_cur);
    k1_zcat<<<(MM * BB) / 256, 256, 0, stream>>>(z, zcatw);
    k2_stream<<<((BB / 2) * 32) / 256, 256, 0, stream>>>(ctx, z_cur, temp1, c0w, c1w);
    k3_gate<<<BB / 8, 256, 0, stream>>>(c0w, c1w, z, conv_w, conv_b,
                                        W1, b1, W2, b2, temp2, w_expT);
    k4_experts<<<BB / BT, 256, 0, stream>>>(Wx, zcatw, w_expT, A, h, z, out);
}